// RegionLoss_5111011082482
// MI455X (gfx1250) — compile-verified
//
#include <hip/hip_runtime.h>
#include <math.h>

// ---------------- constants ----------------
#define NBc   64
#define NAc   5
#define NCc   20
#define NHc   38
#define NWc   38
#define MAXTc 40
#define CHc   (NHc * NWc)        // 1444 elems per channel plane
#define CELLSc (NAc * CHc)       // 7220 cells per batch
#define BLK   256
#define CHUNKS ((CELLSc + BLK - 1) / BLK)   // 29
#define NBLOCKS (NBc * CHUNKS)              // 1856  (== 29 * 64)
#define GTSTRIDE 16                          // floats per GT record

__constant__ float d_AW[NAc] = {1.3221f, 3.19275f, 5.05587f, 9.47112f, 11.2364f};
__constant__ float d_AH[NAc] = {1.73145f, 4.00944f, 8.09892f, 4.84053f, 10.0071f};

typedef __attribute__((ext_vector_type(2))) float v2f;
typedef __attribute__((ext_vector_type(8))) float v8f;

__device__ __forceinline__ float sigmoidf(float z) { return 1.0f / (1.0f + expf(-z)); }

__device__ __forceinline__ float sl1(float d) {
  d = fabsf(d);
  return (d < 1.0f) ? 0.5f * d * d : d - 0.5f;
}

__device__ __forceinline__ float iou_cwh(float ax, float ay, float aw, float ah,
                                         float bx, float by, float bw, float bh) {
  float ax1 = ax - aw * 0.5f, ax2 = ax + aw * 0.5f;
  float ay1 = ay - ah * 0.5f, ay2 = ay + ah * 0.5f;
  float bx1 = bx - bw * 0.5f, bx2 = bx + bw * 0.5f;
  float by1 = by - bh * 0.5f, by2 = by + bh * 0.5f;
  float iw = fmaxf(fminf(ax2, bx2) - fmaxf(ax1, bx1), 0.0f);
  float ih = fmaxf(fminf(ay2, by2) - fmaxf(ay1, by1), 0.0f);
  float inter = iw * ih;
  float uni = aw * ah + bw * bh - inter;
  return inter / uni;
}

// ---------------- Kernel A: per-GT preprocessing ----------------
// One thread per batch; serial over 40 targets (cumprod-style validity).
// GT record layout (16 floats):
// [0]=valid [1]=gx [2]=gy [3]=gw [4]=gh [5]=best_a [6]=gi [7]=gj
// [8]=vtx [9]=vty [10]=vtw [11]=vth [12]=t_iou [13]=gcls [14,15]=pad
__global__ void region_pre(const float* __restrict__ out,
                           const float* __restrict__ target,
                           float* __restrict__ gt) {
  int b = blockIdx.x * blockDim.x + threadIdx.x;
  if (b >= NBc) return;
  const float* tb = target + b * (MAXTc * 5);
  float* gb = gt + b * MAXTc * GTSTRIDE;
  bool running = true;
  for (int t = 0; t < MAXTc; ++t) {
    float gcls = tb[t * 5 + 0];
    float txr = tb[t * 5 + 1], tyr = tb[t * 5 + 2];
    float twr = tb[t * 5 + 3], thr = tb[t * 5 + 4];
    running = running && (txr != 0.0f);
    float gx = txr * (float)NWc, gy = tyr * (float)NHc;
    float gw = twr * (float)NWc, gh = thr * (float)NHc;

    // anchor argmax (first-max on ties, like jnp.argmax)
    int best = 0; float bv = -1.0f, baw = d_AW[0], bah = d_AH[0];
#pragma unroll
    for (int a = 0; a < NAc; ++a) {
      float aw = d_AW[a], ah = d_AH[a];
      float inter = fminf(gw, aw) * fminf(gh, ah);
      float v = inter / (gw * gh + aw * ah - inter);
      if (v > bv) { bv = v; best = a; baw = aw; bah = ah; }
    }
    int gi = (int)gx; gi = gi < 0 ? 0 : (gi > NWc - 1 ? NWc - 1 : gi);
    int gj = (int)gy; gj = gj < 0 ? 0 : (gj > NHc - 1 ? NHc - 1 : gj);

    // gathered pred box at (b, best, gj, gi)
    int base = (b * (NAc * (5 + NCc)) + best * (5 + NCc)) * CHc + gj * NWc + gi;
    float xl = out[base + 0 * CHc];
    float yl = out[base + 1 * CHc];
    float wl = out[base + 2 * CHc];
    float hl = out[base + 3 * CHc];
    float px = sigmoidf(xl) + (float)gi;
    float py = sigmoidf(yl) + (float)gj;
    float pw = expf(wl) * baw;
    float ph = expf(hl) * bah;
    float tiou = iou_cwh(gx, gy, gw, gh, px, py, pw, ph);

    gb[t * GTSTRIDE + 0]  = running ? 1.0f : 0.0f;
    gb[t * GTSTRIDE + 1]  = gx;
    gb[t * GTSTRIDE + 2]  = gy;
    gb[t * GTSTRIDE + 3]  = gw;
    gb[t * GTSTRIDE + 4]  = gh;
    gb[t * GTSTRIDE + 5]  = (float)best;
    gb[t * GTSTRIDE + 6]  = (float)gi;
    gb[t * GTSTRIDE + 7]  = (float)gj;
    gb[t * GTSTRIDE + 8]  = gx - (float)gi;
    gb[t * GTSTRIDE + 9]  = gy - (float)gj;
    gb[t * GTSTRIDE + 10] = logf(gw / baw);
    gb[t * GTSTRIDE + 11] = logf(gh / bah);
    gb[t * GTSTRIDE + 12] = tiou;
    gb[t * GTSTRIDE + 13] = gcls;
    gb[t * GTSTRIDE + 14] = 0.0f;
    gb[t * GTSTRIDE + 15] = 0.0f;
  }
}

// ---------------- Kernel B: fused per-cell loss pass ----------------
__global__ __launch_bounds__(BLK) void region_main(const float* __restrict__ out,
                                                   const float* __restrict__ gt,
                                                   float* __restrict__ pbox,
                                                   float* __restrict__ pcls) {
  __shared__ float sg[MAXTc * GTSTRIDE];
  __shared__ float redb[BLK];
  __shared__ float redc[BLK];

  const int b = blockIdx.y;
  const int tid = threadIdx.x;

  for (int k = tid; k < MAXTc * GTSTRIDE; k += BLK)
    sg[k] = gt[b * MAXTc * GTSTRIDE + k];
  __syncthreads();

  float lbox = 0.0f, lcls = 0.0f;
  int cell = blockIdx.x * BLK + tid;
  if (cell < CELLSc) {
    int a = cell / CHc;
    int rem = cell - a * CHc;
    int j = rem / NWc;
    int i = rem - j * NWc;
    int base = (b * (NAc * (5 + NCc)) + a * (5 + NCc)) * CHc + rem;

    float xl = out[base + 0 * CHc];
    float yl = out[base + 1 * CHc];
    float wl = out[base + 2 * CHc];
    float hl = out[base + 3 * CHc];
    float cl = out[base + 4 * CHc];

    float x = sigmoidf(xl);
    float y = sigmoidf(yl);
    float conf = sigmoidf(cl);
    float aw = d_AW[a], ah = d_AH[a];
    float px = x + (float)i;
    float py = y + (float)j;
    float pw = expf(wl) * aw;
    float ph = expf(hl) * ah;

    // max IoU vs valid GT boxes + last-match scan for the scatter indices
    float cur = 0.0f;
    int match = -1;
    for (int t = 0; t < MAXTc; ++t) {
      const float* g = &sg[t * GTSTRIDE];
      if (g[0] != 0.0f) {
        float v = iou_cwh(g[1], g[2], g[3], g[4], px, py, pw, ph);
        cur = fmaxf(cur, v);
        if ((int)g[5] == a && (int)g[6] == i && (int)g[7] == j) match = t;
      }
    }

    if (match >= 0) {
      const float* g = &sg[match * GTSTRIDE];
      // coord losses (coord_mask == 1 here, 0 elsewhere)
      lbox += 0.5f * (sl1(x - g[8]) + sl1(y - g[9]) + sl1(wl - g[10]) + sl1(hl - g[11]));
      // conf loss with OBJECT_SCALE and tconf = t_iou
      float d = conf - g[12];
      lbox += 0.5f * 5.0f * d * d;
      // focal class loss (online log-sum-exp over 20 logits, no scratch array)
      int tc = (int)g[13]; tc = tc < 0 ? 0 : (tc > NCc - 1 ? NCc - 1 : tc);
      float m = -INFINITY, s = 0.0f, ltc = 0.0f;
#pragma unroll
      for (int c = 0; c < NCc; ++c) {
        float l = out[base + (5 + c) * CHc];
        if (c == tc) ltc = l;
        float nm = fmaxf(m, l);
        s = s * expf(m - nm) + expf(l - nm);
        m = nm;
      }
      float logpt = ltc - (m + logf(s));
      float pt = expf(logpt);
      float om = 1.0f - pt;
      lcls += -(om * om) * logpt;
    } else {
      float cm = (cur > 0.4f) ? 0.0f : 1.0f;  // NOOBJECT_SCALE
      lbox += 0.5f * cm * conf * conf;        // tconf == 0 here
    }
  }

  redb[tid] = lbox;
  redc[tid] = lcls;
  __syncthreads();
  for (int s = BLK / 2; s > 0; s >>= 1) {
    if (tid < s) { redb[tid] += redb[tid + s]; redc[tid] += redc[tid + s]; }
    __syncthreads();
  }
  if (tid == 0) {
    int bf = b * gridDim.x + blockIdx.x;
    pbox[bf] = redb[0];
    pcls[bf] = redc[0];
  }
}

// ---------------- Kernel C: WMMA-accumulated final reduction ----------------
// One full wave32 (EXEC all-ones, as WMMA requires). A = ones(16x4), B = a
// 64-value tile of partial sums; D[m,n] += sum_k B[k,n]. Summing every D
// component gives 16 * (sum of all inputs) regardless of the (k,n) packing.
__device__ __forceinline__ float wmma_sum(const float* __restrict__ data,
                                          int ntiles, int lane, float* lds) {
  v2f ones; ones[0] = 1.0f; ones[1] = 1.0f;
  v8f acc = {};
  for (int t = 0; t < ntiles; ++t) {
    v2f bv;
    bv[0] = data[t * 64 + lane * 2 + 0];
    bv[1] = data[t * 64 + lane * 2 + 1];
    acc = __builtin_amdgcn_wmma_f32_16x16x4_f32(
        /*neg_a=*/false, ones, /*neg_b=*/false, bv,
        /*c_mod=*/(short)0, acc, /*reuse_a=*/false, /*reuse_b=*/false);
  }
  float s = acc[0] + acc[1] + acc[2] + acc[3] + acc[4] + acc[5] + acc[6] + acc[7];
  lds[lane] = s;
  __syncthreads();
  float total = 0.0f;
  if (lane == 0) {
    for (int k = 0; k < 32; ++k) total += lds[k];
    total *= (1.0f / 16.0f);
  }
  __syncthreads();
  return total;  // valid on lane 0
}

__global__ __launch_bounds__(32) void region_finish(const float* __restrict__ pbox,
                                                    const float* __restrict__ pcls,
                                                    float* __restrict__ o) {
  __shared__ float lds[32];
  int lane = threadIdx.x;
  const int ntiles = NBLOCKS / 64;  // 1856 / 64 = 29 exactly
  float box = wmma_sum(pbox, ntiles, lane, lds);
  float cls = wmma_sum(pcls, ntiles, lane, lds);
  if (lane == 0) {
    o[0] = box + cls;  // loss
    o[1] = cls;        // loss_cls
    o[2] = box;        // loss_box
  }
}

// ---------------- launch ----------------
extern "C" void kernel_launch(void* const* d_in, const int* in_sizes, int n_in,
                              void* d_out, int out_size, void* d_ws, size_t ws_size,
                              hipStream_t stream) {
  const float* output = (const float*)d_in[0];
  const float* target = (const float*)d_in[1];
  float* o = (float*)d_out;

  float* ws = (float*)d_ws;
  float* gt   = ws;                                   // 64*40*16 = 40960 floats
  float* pbox = gt + NBc * MAXTc * GTSTRIDE;          // 1856 floats
  float* pcls = pbox + NBLOCKS;                       // 1856 floats

  region_pre<<<1, 64, 0, stream>>>(output, target, gt);
  dim3 grid(CHUNKS, NBc);
  region_main<<<grid, BLK, 0, stream>>>(output, gt, pbox, pcls);
  region_finish<<<1, 32, 0, stream>>>(pbox, pcls, o);
}